// MessagePassing_42992622633778
// MI455X (gfx1250) — compile-verified
//
#include <hip/hip_runtime.h>
#include <cstdint>

#define THREADS 256
#define WAVES_PER_BLOCK (THREADS / 32)
#define D_FEAT 64

// ---------------------------------------------------------------------------
// Zero the output (harness poisons d_out with 0xAA before timing).
// ---------------------------------------------------------------------------
__global__ __launch_bounds__(THREADS) void zero_out_kernel(float4* __restrict__ out, int n4) {
    int i = blockIdx.x * blockDim.x + threadIdx.x;
    if (i < n4) out[i] = make_float4(0.f, 0.f, 0.f, 0.f);
}

// ---------------------------------------------------------------------------
// Scatter-add message passing:
//   out[dst[e]] += x[src[e]] * w[e]
// One wave handles 2 edges per iteration (2 x 64 floats = 32 lanes x float4).
// Gather path: global_load_async_to_lds_b128 (ASYNCcnt), double buffered.
// Scatter path: global_atomic_add_f32 (f32 adds resolved at L2).
// ---------------------------------------------------------------------------
__global__ __launch_bounds__(THREADS) void scatter_mp_kernel(
    const float* __restrict__ x,          // [N, 64]
    const long long* __restrict__ src,    // [E]
    const long long* __restrict__ dst,    // [E]
    const float* __restrict__ ew,         // [E]
    float* __restrict__ out,              // [N, 64]
    int nedges)
{
    // Per-wave staging: 2 buffers x 2 edges x 64 floats = 256 floats (1 KB).
    __shared__ float lds[WAVES_PER_BLOCK * 2 * 2 * D_FEAT];

    const int lane          = threadIdx.x & 31;
    const int wave_in_block = threadIdx.x >> 5;
    const int gwave         = (int)((blockIdx.x * blockDim.x + threadIdx.x) >> 5);
    const int nwaves        = (int)((gridDim.x * blockDim.x) >> 5);

    float* wave_lds = &lds[wave_in_block * (2 * 2 * D_FEAT)];

    // Per-lane LDS byte addresses (generic shared pointer: low 32 bits = LDS offset).
    const uint32_t lds_a0 = (uint32_t)(uintptr_t)(&wave_lds[lane * 4]);
    const uint32_t lds_a1 = lds_a0 + (uint32_t)(2 * D_FEAT * sizeof(float));

    const int npairs = (nedges + 1) >> 1;
    const int sub = lane >> 4;   // which of the 2 edges this lane serves
    const int col = lane & 15;   // float4 column within the 64-float row

    // ---- prologue: issue async gather for first pair into buffer 0 ----
    int p = gwave;
    if (p < npairs) {
        long long e = (long long)2 * p + sub;
        long long s = (e < (long long)nedges) ? src[e] : 0;  // clamp: garbage row 0, unused
        const float* ga = x + s * D_FEAT + col * 4;
        asm volatile("global_load_async_to_lds_b128 %0, %1, off"
                     :: "v"(lds_a0), "v"(ga) : "memory");
    }

    int buf = 0;
    for (; p < npairs; p += nwaves) {
        const int pn = p + nwaves;

        // ---- issue next gather into the other buffer, then wait for current ----
        if (pn < npairs) {
            long long en = (long long)2 * pn + sub;
            long long sn = (en < (long long)nedges) ? src[en] : 0;
            const float* gan = x + sn * D_FEAT + col * 4;
            const uint32_t la = buf ? lds_a0 : lds_a1;
            asm volatile("global_load_async_to_lds_b128 %0, %1, off"
                         :: "v"(la), "v"(gan) : "memory");
            asm volatile("s_wait_asynccnt 1" ::: "memory");  // current buffer done, next in flight
        } else {
            asm volatile("s_wait_asynccnt 0" ::: "memory");
        }

        // ---- compute + scatter for pair p from buffer `buf` ----
        const long long e = (long long)2 * p + sub;
        if (e < (long long)nedges) {
            const float     w = ew[e];
            const long long d = dst[e];
            const float4 v = *reinterpret_cast<const float4*>(
                &wave_lds[buf * (2 * D_FEAT) + lane * 4]);
            float* o = out + d * D_FEAT + col * 4;
            unsafeAtomicAdd(o + 0, v.x * w);
            unsafeAtomicAdd(o + 1, v.y * w);
            unsafeAtomicAdd(o + 2, v.z * w);
            unsafeAtomicAdd(o + 3, v.w * w);
        }
        buf ^= 1;
    }
}

// ---------------------------------------------------------------------------
// Launch
// ---------------------------------------------------------------------------
extern "C" void kernel_launch(void* const* d_in, const int* in_sizes, int n_in,
                              void* d_out, int out_size, void* d_ws, size_t ws_size,
                              hipStream_t stream) {
    const float*     x   = (const float*)d_in[0];
    const long long* ei  = (const long long*)d_in[1];   // [2, E] int64
    const float*     ew  = (const float*)d_in[2];       // [E, 1]
    float*           out = (float*)d_out;

    const int nedges = in_sizes[2];                     // E
    const long long* src = ei;                          // edge_index[0]
    const long long* dst = ei + nedges;                 // edge_index[1]

    // 1) zero the output
    const int n4 = out_size / 4;
    const int zblocks = (n4 + THREADS - 1) / THREADS;
    zero_out_kernel<<<zblocks, THREADS, 0, stream>>>((float4*)out, n4);

    // 2) scatter-add message passing
    const int blocks = 1024;  // 8192 waves -> ~74 edge-pairs per wave
    scatter_mp_kernel<<<blocks, THREADS, 0, stream>>>(x, src, dst, ew, out, nedges);
}